// C4ByteTransformer_62380105007340
// MI455X (gfx1250) — compile-verified
//
#include <hip/hip_runtime.h>
#include <hip/hip_bf16.h>
#include <stddef.h>

// ---------------------------------------------------------------------------
// C4ByteTransformer (carry-propagating byte adder over softmax table lookups)
//
// Roofline (MI455X): FLOPs ~0.8G (negligible); HBM bytes dominate.
//   Phase 1: S_base[4][131072] = X[4x512] @ W1[512 rows]   -> read W1 once (269 MB)
//   Phase 2 (x4, sequential over carry): u = exp(10*(S_base + carry@W1[512:514] - 2.5))
//            out = (u @ W2_sum)/Z, carry' = (u @ W2_carry)/Z -> read W2_sum 4x (537 MB)
//   Total ~815 MB => ~35 us at 23.3 TB/s. Matrix cores are free at this
//   intensity; phase 1 runs on V_WMMA_F32_16X16X4_F32 (M padded 4->16).
//   All global streams use 128-bit accesses; W1 tiles go global->LDS via the
//   gfx1250 async-to-LDS path (GLOBAL_LOAD_ASYNC_TO_LDS_B128, ASYNCcnt).
// ---------------------------------------------------------------------------

#define NK     131072   // NUM_ENTRIES
#define DIN    512      // carry-independent input rows of W1
#define DMODEL 256

typedef float v2f __attribute__((ext_vector_type(2)));
typedef float v4f __attribute__((ext_vector_type(4)));
typedef float v8f __attribute__((ext_vector_type(8)));
typedef int   gv4i __attribute__((vector_size(16)));   // matches builtin param type

#if defined(__has_builtin)
#if __has_builtin(__builtin_amdgcn_global_load_async_to_lds_b128) && \
    __has_builtin(__builtin_amdgcn_s_wait_asynccnt)
#define USE_ASYNC_LDS 1
#endif
#endif

// ---------------------------------------------------------------------------
// Phase 1: S_base[j][k] = sum_{d<512} X[j][d] * W1[d][k],  X = [a_emb | b_emb]
// One wave owns a 16-column tile; a block (8 waves) owns 128 columns.
// W1 staged 16 rows x 128 cols per iteration through LDS; each wave fetches
// one full 512B row per async b128 request (no VGPR round-trip).
// CDNA5 f32 16x16x4 fragment layouts:
//   A (16x4):  VGPR v, lane L -> A[L%16][ (L/16)*2 + v ]
//   B (4x16):  VGPR v, lane L -> B[(L/16)*2 + v][ L%16 ]
//   C (16x16): VGPR r, lane L -> C[ r + 8*(L/16) ][ L%16 ]   (rows 0..3 live
//              in VGPRs 0..3 of lanes 0..15 -> direct store of the 4 steps)
// A rows 4..15 are zero: realized with an unconditional ds_load_b64 from row
// (m&3) times a hoisted 0/1 lane mask -- no EXEC save/restore in the hot loop.
// ---------------------------------------------------------------------------
__global__ __launch_bounds__(256)
void c4bt_scores_wmma(const float* __restrict__ a_emb,
                      const float* __restrict__ b_emb,
                      const float* __restrict__ W1,
                      float* __restrict__ S_base)
{
    __shared__ float Xs[4][520];    // stride 520: 520%64==8 -> conflict-free row indexing
    __shared__ float Bt[16][132];   // stride 132 floats = 528B (16B-aligned rows)

    const int t    = threadIdx.x;
    const int lane = t & 31;
    const int wave = t >> 5;        // 0..7
    const int m    = lane & 15;     // M index / N index inside the 16-wide tile
    const int half = lane >> 4;     // K-half selector
    const int nBlk = blockIdx.x * 128;
    const int nTil = nBlk + wave * 16;

    const int   mrow  = m & 3;                    // always-valid Xs row
    const float amask = (m < 4) ? 1.0f : 0.0f;    // zero rows 4..15 of A

    // Stage X = concat(a_emb, b_emb) rows 0..3.
    for (int i = t; i < 4 * DIN; i += 256) {
        int r = i >> 9;
        int c = i & 511;
        Xs[r][c] = (c < 256) ? a_emb[r * 256 + c] : b_emb[r * 256 + (c - 256)];
    }

    v8f acc = {};
    for (int d0 = 0; d0 < DIN; d0 += 16) {
        __syncthreads();            // previous tile fully consumed
        // Stage 16 rows x 128 cols of W1 as 512 float4s; one wave covers one
        // full row (32 lanes x 16B = 512B contiguous) per request.
#pragma unroll
        for (int i = 0; i < 2; ++i) {
            int idx = t + i * 256;          // 0..511 float4 elements
            int r   = idx >> 5;             // 32 float4 per 128-col row
            int c4  = idx & 31;
            const float* gp = &W1[(size_t)(d0 + r) * NK + nBlk + (c4 << 2)];
            float*       lp = &Bt[r][c4 << 2];
#if defined(USE_ASYNC_LDS)
            __builtin_amdgcn_global_load_async_to_lds_b128(
                (__attribute__((address_space(1))) gv4i*)gp,
                (__attribute__((address_space(3))) gv4i*)lp, 0, 0);
#else
            *(v4f*)lp = *(const v4f*)gp;
#endif
        }
        // Touch the next tile while this one streams (global_prefetch_b8).
        if (d0 + 16 < DIN) {
            int pr = t >> 4, pc = (t & 15) << 3;   // 16 rows x 16 probe points
            __builtin_prefetch(&W1[(size_t)(d0 + 16 + pr) * NK + nBlk + pc], 0, 3);
        }
#if defined(USE_ASYNC_LDS)
        __builtin_amdgcn_s_wait_asynccnt(0);   // my async writes landed in LDS
#endif
        __syncthreads();                       // everyone's writes visible
#pragma unroll
        for (int kk = 0; kk < 4; ++kk) {
            int kb = kk * 4 + half * 2;
            // Unconditional 8B LDS load, masked to zero for A rows 4..15.
            v2f av = *(const v2f*)&Xs[mrow][d0 + kb];
            v2f a  = av * amask;
            v2f b;
            b.x = Bt[kb + 0][wave * 16 + m];
            b.y = Bt[kb + 1][wave * 16 + m];
            // (neg_a, A, neg_b, B, c_mod, C, reuse_a, reuse_b)
            acc = __builtin_amdgcn_wmma_f32_16x16x4_f32(
                      false, a, false, b, (short)0, acc, false, false);
        }
    }

    // Rows 0..3 of the accumulator (lanes 0..15, VGPRs 0..3) are the 4 steps.
    if (lane < 16) {
#pragma unroll
        for (int r = 0; r < 4; ++r)
            S_base[(size_t)r * NK + nTil + lane] = acc[r];
    }
}

// ---------------------------------------------------------------------------
// Phase 2 (per step j): per-block k-chunk of 512 entries.
//   A) u[k] = exp(10*(S_base[j][k] + c0*W1[512][k] + c1*W1[513][k] - 2.5)) -> LDS
//   B) W2_sum streamed with b128: thread t = (g = t>>6, c4 = t&63) owns
//      columns 4*c4..4*c4+3 for k-slice {k : k%4 == g}; per-wave requests are
//      32 lanes x 16B = 512B contiguous. Cross-group sum via LDS (fixed order,
//      deterministic -> no float atomics).
// partial row layout (stride 260): [0..255]=sum cols, [256..257]=carry, [258]=Z
// ---------------------------------------------------------------------------
__global__ __launch_bounds__(256)
void c4bt_step(const float* __restrict__ W1,
               const float* __restrict__ W2_sum,
               const float* __restrict__ W2_carry,
               const float* __restrict__ S_base,
               const float* __restrict__ carry,
               float* __restrict__ partial,
               int j)
{
    __shared__ float u[512];
    __shared__ float redm[4][260];   // per-k-group column partials
    __shared__ float red[256];       // Z tree

    const int    t  = threadIdx.x;
    const int    b  = blockIdx.x;
    const size_t k0 = (size_t)b * 512;
    const int    g  = t >> 6;        // k-group 0..3
    const int    c4 = t & 63;        // float4 column group 0..63

    const float c0 = carry[0];
    const float c1 = carry[1];

#pragma unroll
    for (int i = 0; i < 2; ++i) {
        int    kk = t + i * 256;
        size_t k  = k0 + kk;
        float  s  = S_base[(size_t)j * NK + k]
                  + c0 * W1[(size_t)512 * NK + k]
                  + c1 * W1[(size_t)513 * NK + k];
        u[kk] = __expf((s - 2.5f) * 10.0f);   // bounded inputs: no max pass needed
    }
    __syncthreads();

    // B) rank-1 streaming of W2_sum with 128-bit loads.
    v4f acc = {};
#pragma unroll 4
    for (int kk = g; kk < 512; kk += 4) {
        v4f w = *(const v4f*)&W2_sum[(k0 + kk) * DMODEL + (c4 << 2)];
        acc += u[kk] * w;
    }
    *(v4f*)&redm[g][c4 << 2] = acc;

    if (t < 2) {
        float ca = 0.0f;
        for (int kk = 0; kk < 512; ++kk)
            ca += u[kk] * W2_carry[(k0 + kk) * 2 + t];
        partial[(size_t)b * 260 + 256 + t] = ca;
    }

    red[t] = u[t] + u[t + 256];
    __syncthreads();

    // Column sums across the 4 k-groups (fixed order).
    partial[(size_t)b * 260 + t] =
        redm[0][t] + redm[1][t] + redm[2][t] + redm[3][t];

    // Z partial: fixed-order LDS tree (deterministic).
    for (int s = 128; s > 0; s >>= 1) {
        if (t < s) red[t] += red[t + s];
        __syncthreads();
    }
    if (t == 0) partial[(size_t)b * 260 + 258] = red[0];
}

// Reduce 256 block-partials, normalize by Z, emit out[j][:] and next carry.
__global__ __launch_bounds__(256)
void c4bt_finalize(const float* __restrict__ partial,
                   float* __restrict__ carry,
                   float* __restrict__ out,
                   int j)
{
    __shared__ float sZ, sc0, sc1;
    const int t = threadIdx.x;

    float s = 0.0f;
    for (int b = 0; b < 256; ++b)
        s += partial[(size_t)b * 260 + t];

    if (t == 0) {
        float z = 0.0f, ca = 0.0f, cb = 0.0f;
        for (int b = 0; b < 256; ++b) {
            z  += partial[(size_t)b * 260 + 258];
            ca += partial[(size_t)b * 260 + 256];
            cb += partial[(size_t)b * 260 + 257];
        }
        sZ = z; sc0 = ca; sc1 = cb;
    }
    __syncthreads();

    out[j * DMODEL + t] = s / sZ;
    if (t == 0) { carry[0] = sc0 / sZ; carry[1] = sc1 / sZ; }
}

__global__ void c4bt_init(float* __restrict__ carry)
{
    if (threadIdx.x == 0) { carry[0] = 1.0f; carry[1] = 0.0f; }
}

// ---------------------------------------------------------------------------
extern "C" void kernel_launch(void* const* d_in, const int* in_sizes, int n_in,
                              void* d_out, int out_size, void* d_ws, size_t ws_size,
                              hipStream_t stream)
{
    const float* a_emb    = (const float*)d_in[0];
    const float* b_emb    = (const float*)d_in[1];
    const float* W1       = (const float*)d_in[2];   // [514, 131072]
    const float* W2_sum   = (const float*)d_in[3];   // [131072, 256]
    const float* W2_carry = (const float*)d_in[4];   // [131072, 2]
    float*       out      = (float*)d_out;           // [4, 256]

    float* ws      = (float*)d_ws;
    float* S_base  = ws;                 // 4*NK floats (2 MB)
    float* carry   = ws + 4 * (size_t)NK;
    float* partial = carry + 8;          // 256 blocks * 260 floats (~264 KB)

    c4bt_init<<<1, 32, 0, stream>>>(carry);
    c4bt_scores_wmma<<<dim3(NK / 128), 256, 0, stream>>>(a_emb, b_emb, W1, S_base);

    for (int j = 0; j < 4; ++j) {
        c4bt_step<<<dim3(256), 256, 0, stream>>>(W1, W2_sum, W2_carry,
                                                 S_base, carry, partial, j);
        c4bt_finalize<<<1, 256, 0, stream>>>(partial, carry, out, j);
    }
}